// MAMCA_19834158973381
// MI455X (gfx1250) — compile-verified
//
#include <hip/hip_runtime.h>
#include <hip/hip_bf16.h>
#include <math.h>

// ---------------------------------------------------------------------------
// Problem constants
// ---------------------------------------------------------------------------
#define BB   512
#define LL   128
#define DM   16
#define DI   32
#define NN   16
#define KK   4
#define NLY  4
#define CC   11
#define ROWS (BB * LL)   // 65536

typedef __attribute__((ext_vector_type(2))) float v2f;
typedef __attribute__((ext_vector_type(8))) float v8f;

// ---------------------------------------------------------------------------
// 1) CNN denoiser: two BasicBlocks, 2 -> 16 channels, length preserved.
//    One block per batch element, one thread per sequence position.
// ---------------------------------------------------------------------------
__global__ void __launch_bounds__(128)
denoiser_kernel(const float* __restrict__ x,
                const float* __restrict__ w1, const float* __restrict__ g1, const float* __restrict__ b1,
                const float* __restrict__ w2, const float* __restrict__ g2, const float* __restrict__ b2,
                const float* __restrict__ wd, const float* __restrict__ gd, const float* __restrict__ bd,
                const float* __restrict__ w3, const float* __restrict__ g3, const float* __restrict__ b3,
                const float* __restrict__ w4, const float* __restrict__ g4, const float* __restrict__ b4,
                float* __restrict__ hout)
{
    __shared__ float sw1[96], sw2[768], sw3[768], sw4[768], swd[32];
    __shared__ float sv[10][16];  // g1,b1,g2,b2,gd,bd,g3,b3,g4,b4
    __shared__ float sin0[128], sin1[128];
    __shared__ float bufA[16][128];
    __shared__ float bufB[16][128];

    const int l = threadIdx.x;
    const int b = blockIdx.x;

    for (int i = l; i < 96; i += 128) sw1[i] = w1[i];
    for (int i = l; i < 768; i += 128) { sw2[i] = w2[i]; sw3[i] = w3[i]; sw4[i] = w4[i]; }
    if (l < 32) swd[l] = wd[l];
    if (l < 16) {
        sv[0][l] = g1[l]; sv[1][l] = b1[l]; sv[2][l] = g2[l]; sv[3][l] = b2[l];
        sv[4][l] = gd[l]; sv[5][l] = bd[l]; sv[6][l] = g3[l]; sv[7][l] = b3[l];
        sv[8][l] = g4[l]; sv[9][l] = b4[l];
    }
    sin0[l] = x[b * 256 + l];
    sin1[l] = x[b * 256 + 128 + l];
    __syncthreads();

    // ---- block1 conv1: 2->16, k=3, pad 1, scale, relu
    {
        float v0[3], v1[3];
        #pragma unroll
        for (int k = 0; k < 3; ++k) {
            int ll = l - 1 + k;
            bool ok = (ll >= 0) && (ll < 128);
            v0[k] = ok ? sin0[ll] : 0.f;
            v1[k] = ok ? sin1[ll] : 0.f;
        }
        #pragma unroll
        for (int o = 0; o < 16; ++o) {
            float acc = 0.f;
            #pragma unroll
            for (int k = 0; k < 3; ++k)
                acc += sw1[o * 6 + k] * v0[k] + sw1[o * 6 + 3 + k] * v1[k];
            bufA[o][l] = fmaxf(sv[0][o] * acc + sv[1][o], 0.f);
        }
    }
    __syncthreads();

    // ---- block1 conv2 (16->16) + conv1x1 shortcut (2->16), relu(sum)
    {
        float acc[16];
        #pragma unroll
        for (int o = 0; o < 16; ++o) acc[o] = 0.f;
        #pragma unroll
        for (int i = 0; i < 16; ++i) {
            float v0 = (l > 0)   ? bufA[i][l - 1] : 0.f;
            float v1 = bufA[i][l];
            float v2 = (l < 127) ? bufA[i][l + 1] : 0.f;
            #pragma unroll
            for (int o = 0; o < 16; ++o) {
                const float* wr = &sw2[o * 48 + i * 3];
                acc[o] += wr[0] * v0 + wr[1] * v1 + wr[2] * v2;
            }
        }
        float x0 = sin0[l], x1 = sin1[l];
        #pragma unroll
        for (int o = 0; o < 16; ++o) {
            float sc = sv[4][o] * (swd[o * 2] * x0 + swd[o * 2 + 1] * x1) + sv[5][o];
            bufB[o][l] = fmaxf(sv[2][o] * acc[o] + sv[3][o] + sc, 0.f);
        }
    }
    __syncthreads();

    // ---- block2 conv1 (16->16), scale, relu
    {
        float acc[16];
        #pragma unroll
        for (int o = 0; o < 16; ++o) acc[o] = 0.f;
        #pragma unroll
        for (int i = 0; i < 16; ++i) {
            float v0 = (l > 0)   ? bufB[i][l - 1] : 0.f;
            float v1 = bufB[i][l];
            float v2 = (l < 127) ? bufB[i][l + 1] : 0.f;
            #pragma unroll
            for (int o = 0; o < 16; ++o) {
                const float* wr = &sw3[o * 48 + i * 3];
                acc[o] += wr[0] * v0 + wr[1] * v1 + wr[2] * v2;
            }
        }
        #pragma unroll
        for (int o = 0; o < 16; ++o)
            bufA[o][l] = fmaxf(sv[6][o] * acc[o] + sv[7][o], 0.f);
    }
    __syncthreads();

    // ---- block2 conv2 + identity shortcut, relu; write [B,L,16]
    {
        float acc[16];
        #pragma unroll
        for (int o = 0; o < 16; ++o) acc[o] = 0.f;
        #pragma unroll
        for (int i = 0; i < 16; ++i) {
            float v0 = (l > 0)   ? bufA[i][l - 1] : 0.f;
            float v1 = bufA[i][l];
            float v2 = (l < 127) ? bufA[i][l + 1] : 0.f;
            #pragma unroll
            for (int o = 0; o < 16; ++o) {
                const float* wr = &sw4[o * 48 + i * 3];
                acc[o] += wr[0] * v0 + wr[1] * v1 + wr[2] * v2;
            }
        }
        #pragma unroll
        for (int o = 0; o < 16; ++o) {
            float out = fmaxf(sv[8][o] * acc[o] + sv[9][o] + bufB[o][l], 0.f);
            hout[((size_t)b * 128 + l) * 16 + o] = out;
        }
    }
}

// ---------------------------------------------------------------------------
// 2) residual accumulate + RMSNorm (DM=16 per row).
//    add==0:  res = h;  add==1: res = h + res.  u = rms(res) * w
// ---------------------------------------------------------------------------
__global__ void __launch_bounds__(256)
addnorm_kernel(const float* __restrict__ hin, float* __restrict__ res,
               float* __restrict__ uout, const float* __restrict__ w,
               int add, int nrows)
{
    int row = blockIdx.x * blockDim.x + threadIdx.x;
    if (row >= nrows) return;
    float r[16];
    const float4* hp = (const float4*)(hin + (size_t)row * 16);
    float4* rp = (float4*)(res + (size_t)row * 16);
    #pragma unroll
    for (int q = 0; q < 4; ++q) {
        float4 hv = hp[q];
        if (add) {
            float4 rv = rp[q];
            hv.x += rv.x; hv.y += rv.y; hv.z += rv.z; hv.w += rv.w;
        }
        r[q * 4 + 0] = hv.x; r[q * 4 + 1] = hv.y;
        r[q * 4 + 2] = hv.z; r[q * 4 + 3] = hv.w;
        rp[q] = hv;
    }
    float ss = 0.f;
    #pragma unroll
    for (int c = 0; c < 16; ++c) ss += r[c] * r[c];
    float inv = rsqrtf(ss * (1.f / 16.f) + 1e-5f);
    float4* up = (float4*)(uout + (size_t)row * 16);
    #pragma unroll
    for (int q = 0; q < 4; ++q) {
        float4 uv;
        uv.x = r[q * 4 + 0] * inv * w[q * 4 + 0];
        uv.y = r[q * 4 + 1] * inv * w[q * 4 + 1];
        uv.z = r[q * 4 + 2] * inv * w[q * 4 + 2];
        uv.w = r[q * 4 + 3] * inv * w[q * 4 + 3];
        up[q] = uv;
    }
}

// ---------------------------------------------------------------------------
// 3) Generic C = A @ W.T (+bias) with V_WMMA_F32_16X16X4_F32.
//    A: [M,K] row-major. W: [Nreal,K] row-major. One 16x16 C tile per wave32.
//    A frag  (16x4): lanes 0-15 row m0+lid K={k,k+1}; lanes 16-31 K={k+2,k+3}
//    B frag  (4x16): mirrored (lanes address column n0+lid of W)
//    C layout (v8f): vgpr r -> row m0 + 8*(lane>=16) + r, col n0 + (lane&15)
//
//    Padded-column handling: C column n depends only on B column n, so for
//    n >= Nreal we simply clamp the W pointer to row 0 (valid memory) and
//    let the accumulator hold garbage -- the store is guarded. This keeps
//    the K loop branch-free (no exec manipulation around the WMMA).
// ---------------------------------------------------------------------------
__global__ void __launch_bounds__(256)
gemm_xWt_wmma(const float* __restrict__ A, const float* __restrict__ W,
              float* __restrict__ C, const float* __restrict__ bias,
              int M, int K, int Npad, int Nreal, int ldc)
{
    const int lane = threadIdx.x & 31;
    const int wavesPerBlk = blockDim.x >> 5;
    const int wave = blockIdx.x * wavesPerBlk + (threadIdx.x >> 5);
    const int ntiles = Npad >> 4;
    const int mtiles = M >> 4;
    if (wave >= mtiles * ntiles) return;   // wave-uniform: EXEC stays all-ones

    const int tm = wave / ntiles, tn = wave % ntiles;
    const int m0 = tm << 4, n0 = tn << 4;
    const int half = lane >> 4;      // 0: K pair (k,k+1); 1: (k+2,k+3)
    const int lid  = lane & 15;

    const float* Arow = A + (size_t)(m0 + lid) * K;
    const int    bcol = n0 + lid;
    const float* Wrow = W + (size_t)((bcol < Nreal) ? bcol : 0) * K;

    v8f acc = {};
    #pragma unroll 4
    for (int k = 0; k < K; k += 4) {
        const int ka = k + (half << 1);
        v2f a, bb;
        a.x  = Arow[ka];
        a.y  = Arow[ka + 1];
        bb.x = Wrow[ka];
        bb.y = Wrow[ka + 1];
        acc = __builtin_amdgcn_wmma_f32_16x16x4_f32(
                  false, a, false, bb, (short)0, acc, false, false);
    }

    if (bcol < Nreal) {
        float badd = bias ? bias[bcol] : 0.f;
        int rbase = m0 + (half << 3);
        #pragma unroll
        for (int r = 0; r < 8; ++r)
            C[(size_t)(rbase + r) * ldc + bcol] = acc[r] + badd;
    }
}

// ---------------------------------------------------------------------------
// 4) causal depthwise conv1d (K=4, left pad 3) + bias + SiLU.
//    xz: [B,L,64] (xx = cols 0..31) -> xs: [B,L,32]
// ---------------------------------------------------------------------------
__global__ void __launch_bounds__(256)
dwconv_silu_kernel(const float* __restrict__ xz, const float* __restrict__ cw,
                   const float* __restrict__ cb, float* __restrict__ xs, int total)
{
    int idx = blockIdx.x * blockDim.x + threadIdx.x;
    if (idx >= total) return;
    int d = idx & 31;
    int l = (idx >> 5) & 127;
    int b = idx >> 12;
    float acc = cb[d];
    #pragma unroll
    for (int k = 0; k < 4; ++k) {
        int ll = l - 3 + k;
        if (ll >= 0)
            acc += xz[((size_t)(b * 128 + ll)) * 64 + d] * cw[d * 4 + k];
    }
    float s = acc / (1.f + expf(-acc));  // SiLU
    xs[((size_t)(b * 128 + l)) * 32 + d] = s;
}

// ---------------------------------------------------------------------------
// 5) selective scan: 16 lanes (one per state n) per (b,d) series.
//    xp: [B,L,48] = [dt_in | Bm(16) | Cm(16) | pad]; xs: [B,L,32]; xz: [B,L,64]
//    yg: [B,L,32] = (h.Cm + xs*Dp) * silu(z)
// ---------------------------------------------------------------------------
__global__ void __launch_bounds__(256)
scan_kernel(const float* __restrict__ xp, const float* __restrict__ xs,
            const float* __restrict__ xz, const float* __restrict__ dtw,
            const float* __restrict__ dtb, const float* __restrict__ Al,
            const float* __restrict__ Dpv, float* __restrict__ yg)
{
    int tid = blockIdx.x * blockDim.x + threadIdx.x;
    int n = tid & 15;
    int g = tid >> 4;      // (b,d) pair index
    int d = g & 31;
    int b = g >> 5;

    float dtwd = dtw[d];
    float dtbd = dtb[d];
    float An   = -expf(Al[d * 16 + n]);
    float Dpd  = Dpv[d];

    float h = 0.f;
    size_t rb = (size_t)b * 128;
    for (int l = 0; l < 128; ++l) {
        size_t row = rb + l;
        const float* xpr = xp + row * 48;
        float dtin = xpr[0];
        float Bn   = xpr[1 + n];
        float Cn   = xpr[17 + n];
        float xv   = xs[row * 32 + d];
        float zv   = xz[row * 64 + 32 + d];

        float dt = dtin * dtwd + dtbd;
        dt = (dt > 20.f) ? dt : log1pf(expf(dt));   // softplus
        h = expf(dt * An) * h + dt * Bn * xv;

        float p = h * Cn;                            // einsum partial
        p += __shfl_xor(p, 1, 16);
        p += __shfl_xor(p, 2, 16);
        p += __shfl_xor(p, 4, 16);
        p += __shfl_xor(p, 8, 16);
        if (n == 0) {
            float y  = p + xv * Dpd;
            float gz = zv / (1.f + expf(-zv));       // silu(z)
            yg[row * 32 + d] = y * gz;
        }
    }
}

// ---------------------------------------------------------------------------
// launch
// ---------------------------------------------------------------------------
extern "C" void kernel_launch(void* const* d_in, const int* in_sizes, int n_in,
                              void* d_out, int out_size, void* d_ws, size_t ws_size,
                              hipStream_t stream)
{
    const float* x       = (const float*)d_in[0];
    const float* dn1_w1  = (const float*)d_in[1];
    const float* dn1_g1  = (const float*)d_in[2];
    const float* dn1_b1  = (const float*)d_in[3];
    const float* dn1_w2  = (const float*)d_in[4];
    const float* dn1_g2  = (const float*)d_in[5];
    const float* dn1_b2  = (const float*)d_in[6];
    const float* dn1_wd  = (const float*)d_in[7];
    const float* dn1_gd  = (const float*)d_in[8];
    const float* dn1_bd  = (const float*)d_in[9];
    const float* dn2_w1  = (const float*)d_in[10];
    const float* dn2_g1  = (const float*)d_in[11];
    const float* dn2_b1  = (const float*)d_in[12];
    const float* dn2_w2  = (const float*)d_in[13];
    const float* dn2_g2  = (const float*)d_in[14];
    const float* dn2_b2  = (const float*)d_in[15];
    const float* in_proj = (const float*)d_in[16];  // [NL,64,16]
    const float* conv_w  = (const float*)d_in[17];  // [NL,32,1,4]
    const float* conv_b  = (const float*)d_in[18];  // [NL,32]
    const float* x_proj  = (const float*)d_in[19];  // [NL,33,32]
    const float* dt_w    = (const float*)d_in[20];  // [NL,32,1]
    const float* dt_b    = (const float*)d_in[21];  // [NL,32]
    const float* A_log   = (const float*)d_in[22];  // [NL,32,16]
    const float* Dp      = (const float*)d_in[23];  // [NL,32]
    const float* out_prj = (const float*)d_in[24];  // [NL,16,32]
    const float* norm_w  = (const float*)d_in[25];  // [NL,16]
    const float* normf_w = (const float*)d_in[26];  // [16]
    const float* fc_w    = (const float*)d_in[27];  // [11,2048]
    const float* fc_b    = (const float*)d_in[28];  // [11]

    float* ws  = (float*)d_ws;
    float* h   = ws;                  // ROWS*16  = 1,048,576
    float* res = h  + (size_t)ROWS * 16;
    float* u   = res + (size_t)ROWS * 16;
    float* xz  = u  + (size_t)ROWS * 16;   // ROWS*64 = 4,194,304
    float* xs  = xz + (size_t)ROWS * 64;   // ROWS*32
    float* xpj = xs + (size_t)ROWS * 32;   // ROWS*48
    float* yg  = xpj + (size_t)ROWS * 48;  // ROWS*32

    denoiser_kernel<<<BB, 128, 0, stream>>>(
        x, dn1_w1, dn1_g1, dn1_b1, dn1_w2, dn1_g2, dn1_b2,
        dn1_wd, dn1_gd, dn1_bd, dn2_w1, dn2_g1, dn2_b1,
        dn2_w2, dn2_g2, dn2_b2, h);

    for (int i = 0; i < NLY; ++i) {
        // res (+)= h ; u = rmsnorm(res)*norm_w[i]
        addnorm_kernel<<<ROWS / 256, 256, 0, stream>>>(
            h, res, u, norm_w + i * 16, (i > 0) ? 1 : 0, ROWS);
        // xz = u @ in_proj[i].T    M=65536 K=16 N=64
        gemm_xWt_wmma<<<(4096 * 4) / 8, 256, 0, stream>>>(
            u, in_proj + (size_t)i * 64 * 16, xz, nullptr, ROWS, 16, 64, 64, 64);
        // xs = silu(depthwise_conv(xz[:, :32]))
        dwconv_silu_kernel<<<(ROWS * DI) / 256, 256, 0, stream>>>(
            xz, conv_w + (size_t)i * DI * KK, conv_b + (size_t)i * DI, xs, ROWS * DI);
        // xp = xs @ x_proj[i].T    M=65536 K=32 N=33 (pad 48)
        gemm_xWt_wmma<<<(4096 * 3) / 8, 256, 0, stream>>>(
            xs, x_proj + (size_t)i * 33 * 32, xpj, nullptr, ROWS, 32, 48, 33, 48);
        // selective scan + gate
        scan_kernel<<<(BB * DI * NN) / 256, 256, 0, stream>>>(
            xpj, xs, xz, dt_w + (size_t)i * DI, dt_b + (size_t)i * DI,
            A_log + (size_t)i * DI * NN, Dp + (size_t)i * DI, yg);
        // h = yg @ out_proj[i].T   M=65536 K=32 N=16
        gemm_xWt_wmma<<<4096 / 8, 256, 0, stream>>>(
            yg, out_prj + (size_t)i * 16 * 32, h, nullptr, ROWS, 32, 16, 16, 16);
    }

    // final: res = h + res ; u = rmsnorm(res)*norm_f_w  (u == [B, L*16] flat)
    addnorm_kernel<<<ROWS / 256, 256, 0, stream>>>(h, res, u, normf_w, 1, ROWS);

    // logits = u @ fc_w.T + fc_b   M=512 K=2048 N=11 (pad 16)
    gemm_xWt_wmma<<<4, 256, 0, stream>>>(
        u, fc_w, (float*)d_out, fc_b, BB, 2048, 16, CC, CC);
}